// EndoMamba_14894946583155
// MI455X (gfx1250) — compile-verified
//
#include <hip/hip_runtime.h>
#include <hip/hip_bf16.h>
#include <math.h>

// ---------------------------------------------------------------------------
// EndoMamba forward on MI455X (gfx1250).
// GEMMs: V_WMMA_F32_16X16X4_F32, 2x2 register-tiled (32x32 per wave),
// branch-free inner loop via index clamping (OOB rows feed unstored outputs).
// SSM scan: one lane per (sequence, channel), 16-wide state in VGPRs.
// ---------------------------------------------------------------------------

#define Dm    192
#define DIm   384
#define DSm   16
#define KCm   4
#define DTRm  12
#define XPm   44          // DTR + 2*DS
#define NCLSm 1000
#define Bm    4
#define Tm    8
#define NPm   196
#define Mtok  6272        // B*T*NP
#define L_SP  196
#define NS_SP 32
#define L_TM  1568
#define NS_TM 4

#define CDIV(a,b) (((a)+(b)-1)/(b))

typedef __attribute__((ext_vector_type(2))) float v2f;
typedef __attribute__((ext_vector_type(8))) float v8f;

__device__ __forceinline__ v8f wmma4(v2f a, v2f b, v8f c)
{
    return __builtin_amdgcn_wmma_f32_16x16x4_f32(
        false, a, false, b, (short)0, c, false, false);
}

// ---------------------------------------------------------------------------
// C[M,N] = A[M,K(lda)] * W[N,K]^T  (+bias / softplus)
// One 32x32 output supertile per wave (2x2 WMMA tiles), 4 waves per block.
// Clamped addressing: no divergence in the K loop; OOB products land only in
// output elements that the store guards discard.
// epi: 0 = none, 1 = +bias, 2 = softplus(x + bias)
// ---------------------------------------------------------------------------
__global__ void em_gemm_wmma(const float* __restrict__ A, int lda,
                             const float* __restrict__ W,
                             const float* __restrict__ bias,
                             float* __restrict__ C,
                             int Mr, int Nr, int K, int epi)
{
    const int wave   = threadIdx.x >> 5;
    const int lane   = threadIdx.x & 31;
    const int tilesN = (Nr + 31) >> 5;          // 32-wide supertiles
    const int tileId = blockIdx.x * 4 + wave;
    const int mt = tileId / tilesN;
    const int nt = tileId % tilesN;
    if (mt * 32 >= Mr) return;                  // uniform per wave

    const int half = lane >> 4;                 // selects K pair {0,1} vs {2,3}
    const int ml   = lane & 15;

    // clamp: OOB rows compute garbage that is never stored
    int ar0 = mt * 32 + ml;       if (ar0 >= Mr) ar0 = Mr - 1;
    int ar1 = mt * 32 + 16 + ml;  if (ar1 >= Mr) ar1 = Mr - 1;
    int wr0 = nt * 32 + ml;       if (wr0 >= Nr) wr0 = Nr - 1;
    int wr1 = nt * 32 + 16 + ml;  if (wr1 >= Nr) wr1 = Nr - 1;

    const float* Ap0 = A + (size_t)ar0 * lda + 2 * half;
    const float* Ap1 = A + (size_t)ar1 * lda + 2 * half;
    const float* Wp0 = W + (size_t)wr0 * K   + 2 * half;
    const float* Wp1 = W + (size_t)wr1 * K   + 2 * half;

    v8f c00 = {}, c01 = {}, c10 = {}, c11 = {};
    for (int k = 0; k < K; k += 4) {
        v2f a0 = *(const v2f*)(Ap0 + k);
        v2f a1 = *(const v2f*)(Ap1 + k);
        v2f b0 = *(const v2f*)(Wp0 + k);
        v2f b1 = *(const v2f*)(Wp1 + k);
        c00 = wmma4(a0, b0, c00);
        c01 = wmma4(a0, b1, c01);
        c10 = wmma4(a1, b0, c10);
        c11 = wmma4(a1, b1, c11);
    }

    const int col0 = nt * 32 + ml;
    const int col1 = nt * 32 + 16 + ml;
    float bv0 = 0.f, bv1 = 0.f;
    if (epi != 0 && bias != nullptr) {
        if (col0 < Nr) bv0 = bias[col0];
        if (col1 < Nr) bv1 = bias[col1];
    }
#pragma unroll
    for (int r = 0; r < 8; ++r) {
        const int row0 = mt * 32 + r + 8 * half;
        const int row1 = row0 + 16;
        float v;
        if (row0 < Mr && col0 < Nr) {
            v = c00[r] + bv0;
            if (epi == 2) v = (v > 20.f) ? v : log1pf(__expf(v));
            C[(size_t)row0 * Nr + col0] = v;
        }
        if (row0 < Mr && col1 < Nr) {
            v = c01[r] + bv1;
            if (epi == 2) v = (v > 20.f) ? v : log1pf(__expf(v));
            C[(size_t)row0 * Nr + col1] = v;
        }
        if (row1 < Mr && col0 < Nr) {
            v = c10[r] + bv0;
            if (epi == 2) v = (v > 20.f) ? v : log1pf(__expf(v));
            C[(size_t)row1 * Nr + col0] = v;
        }
        if (row1 < Mr && col1 < Nr) {
            v = c11[r] + bv1;
            if (epi == 2) v = (v > 20.f) ? v : log1pf(__expf(v));
            C[(size_t)row1 * Nr + col1] = v;
        }
    }
}

// ---------------------------------------------------------------------------
// Patch gather: x(B,3,T,224,224) -> A(M,768) with f = c*256 + py*16 + px
// ---------------------------------------------------------------------------
__global__ void em_patch_gather(const float* __restrict__ x,
                                float* __restrict__ A, int total)
{
    int idx = blockIdx.x * blockDim.x + threadIdx.x;
    if (idx >= total) return;
    int f  = idx % 768;
    int m  = idx / 768;
    int np = m % NPm;
    int bt = m / NPm;
    int t  = bt % Tm;
    int b  = bt / Tm;
    int ph = np / 14, pw = np % 14;
    int c  = f / 256;
    int r  = f % 256;
    int py = r / 16, px = r % 16;
    size_t xi = ((((size_t)b * 3 + c) * Tm + t) * 224 + (ph * 16 + py)) * 224
              + (pw * 16 + px);
    A[idx] = x[xi];
}

// tok += pos_embed[np,d] + temporal sinusoid(t,d)
__global__ void em_add_pos_tpe(float* __restrict__ tok,
                               const float* __restrict__ pos, int total)
{
    int idx = blockIdx.x * blockDim.x + threadIdx.x;
    if (idx >= total) return;
    int d  = idx % Dm;
    int m  = idx / Dm;
    int np = m % NPm;
    int t  = (m / NPm) % Tm;
    float div = __expf(-9.210340371976184f * (float)(2 * (d / 2)) / (float)Dm);
    float ang = (float)t * div;
    float tpe = (d & 1) ? __cosf(ang) : __sinf(ang);
    tok[idx] += pos[np * Dm + d] + tpe;
}

__global__ void em_zero(float* __restrict__ p, int n)
{
    int i = blockIdx.x * blockDim.x + threadIdx.x;
    if (i < n) p[i] = 0.f;
}

__global__ void em_add(float* __restrict__ dst, const float* __restrict__ src, int n)
{
    int i = blockIdx.x * blockDim.x + threadIdx.x;
    if (i < n) dst[i] += src[i];
}

// ---------------------------------------------------------------------------
// RMSNorm: one wave32 per row of 192, 6 elems/lane, shfl_xor reduction
// ---------------------------------------------------------------------------
__global__ void em_rmsnorm(const float* __restrict__ X,
                           const float* __restrict__ w,
                           float* __restrict__ Y, int Mr)
{
    int row  = blockIdx.x * (blockDim.x >> 5) + (threadIdx.x >> 5);
    int lane = threadIdx.x & 31;
    if (row >= Mr) return;
    const float* xp = X + (size_t)row * Dm;
    float v[6];
    float ss = 0.f;
#pragma unroll
    for (int i = 0; i < 6; ++i) {
        v[i] = xp[lane + 32 * i];
        ss = fmaf(v[i], v[i], ss);
    }
#pragma unroll
    for (int off = 16; off > 0; off >>= 1)
        ss += __shfl_xor(ss, off, 32);
    float scale = rsqrtf(ss * (1.f / (float)Dm) + 1e-5f);
#pragma unroll
    for (int i = 0; i < 6; ++i)
        Y[(size_t)row * Dm + lane + 32 * i] = v[i] * scale * w[lane + 32 * i];
}

// ---------------------------------------------------------------------------
// Causal depthwise conv (K=4) + SiLU. rev=1 processes sequence reversed
// (backward branch); xs is stored in scan (logical) time order.
// x lives in xz rows of width 768, channels [0,384).
// ---------------------------------------------------------------------------
__global__ void em_conv_silu(const float* __restrict__ xz,
                             const float* __restrict__ cw,
                             const float* __restrict__ cb,
                             float* __restrict__ xs,
                             int Lr, int rev, int total)
{
    int idx = blockIdx.x * blockDim.x + threadIdx.x;
    if (idx >= total) return;
    int ch  = idx % DIm;
    int tt  = (idx / DIm) % Lr;
    int seq = idx / (DIm * Lr);
    float acc = cb[ch];
#pragma unroll
    for (int j = 0; j < KCm; ++j) {
        int tl = tt - (KCm - 1) + j;
        if (tl >= 0) {
            int tp = rev ? (Lr - 1 - tl) : tl;
            acc = fmaf(xz[((size_t)seq * Lr + tp) * 768 + ch], cw[ch * KCm + j], acc);
        }
    }
    float sig = 1.f / (1.f + __expf(-acc));
    xs[idx] = acc * sig;                       // silu; idx == (seq*Lr+tt)*DI+ch
}

// ---------------------------------------------------------------------------
// Selective scan. One block per sequence, one lane per channel (384 threads,
// 12 wave32s). 16-wide state in VGPRs; 16 exp + 32 fma per step.
// rev=1: output written flipped; accum=1: y += (bidirectional merge).
// ---------------------------------------------------------------------------
__global__ void em_scan(const float* __restrict__ xs,
                        const float* __restrict__ dtb,
                        const float* __restrict__ dbl,
                        const float* __restrict__ A_log,
                        const float* __restrict__ Dssm,
                        float* __restrict__ y,
                        int Lr, int rev, int accum)
{
    const int seq = blockIdx.x;
    const int ch  = threadIdx.x;
    float a[DSm];
#pragma unroll
    for (int s = 0; s < DSm; ++s) a[s] = -__expf(A_log[ch * DSm + s]);
    const float dv = Dssm[ch];
    float h[DSm] = {0.f};
    for (int t = 0; t < Lr; ++t) {
        const size_t m  = (size_t)seq * Lr + t;
        const float dt  = dtb[m * DIm + ch];
        const float xv  = xs[m * DIm + ch];
        const float du  = dt * xv;
        const float* bc = dbl + m * XPm + DTRm;     // B[0..15], C at +16
        float acc = 0.f;
#pragma unroll
        for (int s = 0; s < DSm; ++s) {
            h[s] = fmaf(h[s], __expf(dt * a[s]), du * bc[s]);
            acc  = fmaf(h[s], bc[DSm + s], acc);
        }
        const int tp = rev ? (Lr - 1 - t) : t;
        const size_t mo = ((size_t)seq * Lr + tp) * DIm + ch;
        const float out = acc + dv * xv;
        if (accum) y[mo] += out; else y[mo] = out;
        if (t + 1 < Lr)
            __builtin_prefetch(dbl + (m + 1) * XPm, 0, 0);  // global_prefetch_b8
    }
}

// g = y * silu(z), z = xz[:, 384:768]
__global__ void em_gate(const float* __restrict__ y,
                        const float* __restrict__ xz,
                        float* __restrict__ g, int total)
{
    int idx = blockIdx.x * blockDim.x + threadIdx.x;
    if (idx >= total) return;
    int ch = idx % DIm;
    int m  = idx / DIm;
    float z = xz[(size_t)m * 768 + DIm + ch];
    g[idx] = y[idx] * (z / (1.f + __expf(-z)));
}

// pooled[b,d] = mean_t hf[b,t,d] over 1568 tokens
__global__ void em_pool(const float* __restrict__ hf, float* __restrict__ pooled)
{
    int b = blockIdx.x;
    int d = threadIdx.x;
    float s = 0.f;
    for (int t = 0; t < L_TM; ++t)
        s += hf[((size_t)b * L_TM + t) * Dm + d];
    pooled[b * Dm + d] = s * (1.f / (float)L_TM);
}

// ---------------------------------------------------------------------------
// Host orchestration
// ---------------------------------------------------------------------------
static inline void gemm(const float* A, int lda, const float* W,
                        const float* bias, float* C,
                        int Mr, int Nr, int K, int epi, hipStream_t s)
{
    int tiles = CDIV(Mr, 32) * CDIV(Nr, 32);
    em_gemm_wmma<<<CDIV(tiles, 4), 128, 0, s>>>(A, lda, W, bias, C, Mr, Nr, K, epi);
}

extern "C" void kernel_launch(void* const* d_in, const int* in_sizes, int n_in,
                              void* d_out, int out_size, void* d_ws, size_t ws_size,
                              hipStream_t stream)
{
    const float* x        = (const float*)d_in[0];
    const float* patch_w  = (const float*)d_in[1];
    const float* patch_b  = (const float*)d_in[2];
    const float* pos_emb  = (const float*)d_in[3];
    const float* in_w     = (const float*)d_in[4];
    const float* conv_w   = (const float*)d_in[5];
    const float* conv_b   = (const float*)d_in[6];
    const float* xproj_w  = (const float*)d_in[7];
    const float* dt_w     = (const float*)d_in[8];
    const float* dt_b     = (const float*)d_in[9];
    const float* A_log    = (const float*)d_in[10];
    const float* D_ssm    = (const float*)d_in[11];
    const float* out_w    = (const float*)d_in[12];
    const float* conv_w_b = (const float*)d_in[13];
    const float* conv_b_b = (const float*)d_in[14];
    const float* xproj_wb = (const float*)d_in[15];
    const float* dt_w_b   = (const float*)d_in[16];
    const float* dt_b_b   = (const float*)d_in[17];
    const float* A_log_b  = (const float*)d_in[18];
    const float* D_ssm_b  = (const float*)d_in[19];
    const float* norm_w   = (const float*)d_in[20];
    const float* norm_f_w = (const float*)d_in[21];
    const float* head_w   = (const float*)d_in[22];
    const float* head_b   = (const float*)d_in[23];
    (void)in_sizes; (void)n_in; (void)out_size; (void)ws_size;

    float* ws = (float*)d_ws;
    size_t off = 0;
    auto alloc = [&](size_t n) { float* p = ws + off; off += (n + 63) & ~(size_t)63; return p; };
    float* hbuf   = alloc((size_t)Mtok * Dm);
    float* res    = alloc((size_t)Mtok * Dm);
    float* hn     = alloc((size_t)Mtok * Dm);
    float* xz     = alloc((size_t)Mtok * 768);   // also patch-gather staging
    float* xsb    = alloc((size_t)Mtok * DIm);
    float* dbl    = alloc((size_t)Mtok * XPm);
    float* dtb    = alloc((size_t)Mtok * DIm);
    float* yb     = alloc((size_t)Mtok * DIm);
    float* gb     = alloc((size_t)Mtok * DIm);
    float* pooled = alloc((size_t)Bm * Dm);

    const int nTok = Mtok * Dm;      // 1.2M
    const int nDI  = Mtok * DIm;     // 2.4M

    // ---- patch embed: gather -> WMMA GEMM -> +bias +pos +temporal PE ----
    em_patch_gather<<<CDIV(Mtok * 768, 256), 256, 0, stream>>>(x, xz, Mtok * 768);
    gemm(xz, 768, patch_w, patch_b, hbuf, Mtok, Dm, 768, 1, stream);
    em_add_pos_tpe<<<CDIV(nTok, 256), 256, 0, stream>>>(hbuf, pos_emb, nTok);
    em_zero<<<CDIV(nTok, 256), 256, 0, stream>>>(res, nTok);

    for (int i = 0; i < 8; ++i) {
        const int Lr = (i < 4) ? L_SP : L_TM;
        const int NS = (i < 4) ? NS_SP : NS_TM;

        // res += h ; hn = rmsnorm(res, norm_w[i])
        em_add<<<CDIV(nTok, 256), 256, 0, stream>>>(res, hbuf, nTok);
        em_rmsnorm<<<CDIV(Mtok, 8), 256, 0, stream>>>(res, norm_w + i * Dm, hn, Mtok);

        // in_proj: xz = hn @ in_w[i].T   (M x 768)
        gemm(hn, Dm, in_w + (size_t)i * 768 * Dm, nullptr, xz, Mtok, 768, Dm, 0, stream);

        // ---- forward SSM branch ----
        em_conv_silu<<<CDIV(nDI, 256), 256, 0, stream>>>(
            xz, conv_w + (size_t)i * DIm * KCm, conv_b + (size_t)i * DIm,
            xsb, Lr, 0, nDI);
        gemm(xsb, DIm, xproj_w + (size_t)i * XPm * DIm, nullptr, dbl, Mtok, XPm, DIm, 0, stream);
        gemm(dbl, XPm, dt_w + (size_t)i * DIm * DTRm, dt_b + (size_t)i * DIm,
             dtb, Mtok, DIm, DTRm, 2, stream);
        em_scan<<<NS, DIm, 0, stream>>>(
            xsb, dtb, dbl, A_log + (size_t)i * DIm * DSm, D_ssm + (size_t)i * DIm,
            yb, Lr, 0, 0);

        // ---- backward SSM branch (spatial blocks only) ----
        if (i < 4) {
            em_conv_silu<<<CDIV(nDI, 256), 256, 0, stream>>>(
                xz, conv_w_b + (size_t)i * DIm * KCm, conv_b_b + (size_t)i * DIm,
                xsb, Lr, 1, nDI);
            gemm(xsb, DIm, xproj_wb + (size_t)i * XPm * DIm, nullptr, dbl, Mtok, XPm, DIm, 0, stream);
            gemm(dbl, XPm, dt_w_b + (size_t)i * DIm * DTRm, dt_b_b + (size_t)i * DIm,
                 dtb, Mtok, DIm, DTRm, 2, stream);
            em_scan<<<NS, DIm, 0, stream>>>(
                xsb, dtb, dbl, A_log_b + (size_t)i * DIm * DSm, D_ssm_b + (size_t)i * DIm,
                yb, Lr, 1, 1);
        }

        // gate + out_proj: h = (y * silu(z)) @ out_w[i].T
        em_gate<<<CDIV(nDI, 256), 256, 0, stream>>>(yb, xz, gb, nDI);
        gemm(gb, DIm, out_w + (size_t)i * Dm * DIm, nullptr, hbuf, Mtok, Dm, DIm, 0, stream);
    }

    // final residual, norm, pool, head
    em_add<<<CDIV(nTok, 256), 256, 0, stream>>>(res, hbuf, nTok);
    em_rmsnorm<<<CDIV(Mtok, 8), 256, 0, stream>>>(res, norm_f_w, hn, Mtok);
    em_pool<<<Bm, Dm, 0, stream>>>(hn, pooled);
    gemm(pooled, Dm, head_w, head_b, (float*)d_out, Bm, NCLSm, Dm, 1, stream);
}